// MyMaxPool2D_5093831213629
// MI455X (gfx1250) — compile-verified
//
#include <hip/hip_runtime.h>

// Max-pool 2x2, stride 2, VALID, NCHW fp32.
// Input  : (16, 64, 512, 512)  -> 1.074 GB
// Output : (16, 64, 256, 256)  -> 268 MB
// Pure streaming kernel: ~1.34 GB total traffic, floor ~58 us @ 23.3 TB/s.
//
// Per lane: 4x global_load_b128 (two adjacent input rows x 8 cols),
//           12x f32 max (fused to v_max3/v_max by backend),
//           1x global_store_b128. All accesses 16B-aligned, fully coalesced.
// Non-temporal hints on loads+stores: zero reuse, keep it out of L2.

typedef float f32x4 __attribute__((ext_vector_type(4)));

#define IN_W   512
#define IN_H   512
#define OUT_W  256
#define OUT_H  256
#define OW4    (OUT_W / 4)   // 64 float4-groups per output row

__global__ __launch_bounds__(256) void maxpool2x2_nt_kernel(
    const float* __restrict__ x, float* __restrict__ out, int n_vec) {
  int tid = blockIdx.x * blockDim.x + threadIdx.x;
  if (tid >= n_vec) return;

  // tid indexes one float4 of output: [nc][oy][ox4]
  int ox4 = tid & (OW4 - 1);        // 0..63
  int t   = tid >> 6;
  int oy  = t & (OUT_H - 1);        // 0..255
  int nc  = t >> 8;                 // 0..1023

  const float* row0 = x + ((size_t)nc * IN_H + (size_t)(oy << 1)) * IN_W
                        + (size_t)(ox4 << 3);           // 8 input cols per lane
  const float* row1 = row0 + IN_W;

  f32x4 a0 = __builtin_nontemporal_load((const f32x4*)(row0));
  f32x4 a1 = __builtin_nontemporal_load((const f32x4*)(row0 + 4));
  f32x4 b0 = __builtin_nontemporal_load((const f32x4*)(row1));
  f32x4 b1 = __builtin_nontemporal_load((const f32x4*)(row1 + 4));

  f32x4 r;
  r.x = fmaxf(fmaxf(a0.x, a0.y), fmaxf(b0.x, b0.y));
  r.y = fmaxf(fmaxf(a0.z, a0.w), fmaxf(b0.z, b0.w));
  r.z = fmaxf(fmaxf(a1.x, a1.y), fmaxf(b1.x, b1.y));
  r.w = fmaxf(fmaxf(a1.z, a1.w), fmaxf(b1.z, b1.w));

  __builtin_nontemporal_store(r, (f32x4*)(out + (size_t)tid * 4));
}

extern "C" void kernel_launch(void* const* d_in, const int* in_sizes, int n_in,
                              void* d_out, int out_size, void* d_ws, size_t ws_size,
                              hipStream_t stream) {
  (void)in_sizes; (void)n_in; (void)d_ws; (void)ws_size;
  const float* x = (const float*)d_in[0];
  float* out     = (float*)d_out;

  int n_vec   = out_size / 4;                 // 16,777,216 float4 outputs
  int threads = 256;                          // 8 wave32 per block
  int blocks  = (n_vec + threads - 1) / threads;

  maxpool2x2_nt_kernel<<<blocks, threads, 0, stream>>>(x, out, n_vec);
}